// CompGCNLayer_15006615732839
// MI455X (gfx1250) — compile-verified
//
#include <hip/hip_runtime.h>

// CDNA5 wave32 WMMA vector types
typedef __attribute__((ext_vector_type(2))) float v2f; // A/B frag of 16x16x4 f32
typedef __attribute__((ext_vector_type(8))) float v8f; // C/D 16x16 f32

#define D        128
#define PITCH    132   // LDS row pitch in floats; 132 % 64 != 0 -> spreads banks
#define HALF_REL 50
#define MT       4     // 16-row M-tiles per block (64 rows/block) -> 4x B-fragment reuse

// ---------------------------------------------------------------------------
// out = x @ w_loop   (also initializes d_out)
// block = 256 threads = 8 wave32; block tile = 64 rows x 128 cols.
// Wave w owns cols [16w,16w+16) and keeps MT=4 accumulators so each B
// fragment (2 global_load_b32) feeds 4 v_wmma_f32_16x16x4_f32 ops.
// ---------------------------------------------------------------------------
__global__ __launch_bounds__(256) void selfloop_gemm(
    const float* __restrict__ x, const float* __restrict__ w,
    float* __restrict__ out, int nrows)
{
  __shared__ float xs[MT * 16 * PITCH];   // 64 x 132 f32 = 33.8 KB

  const int t    = threadIdx.x;
  const int lane = t & 31;
  const int wave = t >> 5;
  const int row0 = blockIdx.x * (MT * 16);

  // stage 64x128 A tile (clamped loads; tail-safe)
  for (int idx = t; idx < MT * 16 * D; idx += 256) {
    int r = idx >> 7, c = idx & 127;
    int rg = row0 + r; if (rg >= nrows) rg = nrows - 1;
    xs[r * PITCH + c] = x[(size_t)rg * D + c];
  }
  __syncthreads();

  // WMMA lane mapping (ISA 7.12.2):
  // A 16x4: lane m=lane&15, k-pair=(lane>>4)*2 ; B 4x16: lane n=lane&15, same k-pair
  const int m       = lane & 15;
  const int koff    = (lane >> 4) * 2;
  const int n       = lane & 15;
  const int colbase = wave * 16;

  v8f acc[MT] = {};
  for (int k = 0; k < D; k += 4) {
    v2f b;
    b.x = w[(size_t)(k + koff)     * D + colbase + n];
    b.y = w[(size_t)(k + koff + 1) * D + colbase + n];
    #pragma unroll
    for (int mt = 0; mt < MT; ++mt) {
      v2f a;
      a.x = xs[(mt * 16 + m) * PITCH + k + koff];
      a.y = xs[(mt * 16 + m) * PITCH + k + koff + 1];
      acc[mt] = __builtin_amdgcn_wmma_f32_16x16x4_f32(false, a, false, b,
                                                      (short)0, acc[mt], false, false);
    }
  }

  // C/D layout: VGPR v -> row = v + (lane>>4)*8
  #pragma unroll
  for (int mt = 0; mt < MT; ++mt) {
    const int rbase = row0 + mt * 16;
    if (rbase + 16 <= nrows) {                      // fast path: whole tile in range
      #pragma unroll
      for (int v = 0; v < 8; ++v) {
        int m2 = v + (lane >> 4) * 8;
        out[(size_t)(rbase + m2) * D + colbase + n] = acc[mt][v];
      }
    } else {                                        // partial tail tile
      #pragma unroll
      for (int v = 0; v < 8; ++v) {
        int m2 = v + (lane >> 4) * 8;
        if (rbase + m2 < nrows)
          out[(size_t)(rbase + m2) * D + colbase + n] = acc[mt][v];
      }
    }
  }
}

// ---------------------------------------------------------------------------
// Edge messages: 64 edges per block (4 M-tiles).
// comp = x[src] * rel_emb[rel_idx]; masked into comp_f / comp_b so the
// direction-dependent weight select is one uniform dual-K GEMM:
//   msg = comp_f @ Wf + comp_b @ Wb
// Each k-step: 4 B-element loads feed 8 WMMA ops (4 tiles x 2 chains).
// Rows scaled by edge_norm, then global_atomic_add_f32 scatter into out[dst].
// ---------------------------------------------------------------------------
__global__ __launch_bounds__(256) void edge_msg(
    const float* __restrict__ x,  const float* __restrict__ wf,
    const float* __restrict__ wb, const float* __restrict__ rel,
    const float* __restrict__ enorm, const int* __restrict__ eidx,
    const int* __restrict__ etype, float* __restrict__ out, int nedges)
{
  __shared__ float cf[MT * 16 * PITCH];   // 33.8 KB
  __shared__ float cb[MT * 16 * PITCH];   // 33.8 KB
  __shared__ int   s_dst[MT * 16];
  __shared__ float s_norm[MT * 16];
  __shared__ int   s_src[MT * 16];
  __shared__ int   s_rel[MT * 16];
  __shared__ int   s_fwd[MT * 16];

  const int t    = threadIdx.x;
  const int lane = t & 31;
  const int wave = t >> 5;
  const int e0   = blockIdx.x * (MT * 16);

  if (t < MT * 16) {
    int eg = e0 + t;
    int valid = (eg < nedges);
    int e  = valid ? eg : (nedges - 1);
    int ty = etype[e];
    int f  = (ty < HALF_REL);
    s_fwd[t]  = f;
    s_rel[t]  = f ? ty : ty - HALF_REL;
    s_src[t]  = eidx[e];                  // edge_index row 0 (src)
    s_dst[t]  = eidx[nedges + e];         // edge_index row 1 (dst)
    s_norm[t] = valid ? enorm[e] : 0.0f;  // tail edges contribute 0
  }
  __syncthreads();

  // gather + compose + direction mask (coalesced over D)
  for (int idx = t; idx < MT * 16 * D; idx += 256) {
    int r = idx >> 7, c = idx & 127;
    float comp = x[(size_t)s_src[r] * D + c] * rel[(size_t)s_rel[r] * D + c];
    int f = s_fwd[r];
    cf[r * PITCH + c] = f ? comp : 0.0f;
    cb[r * PITCH + c] = f ? 0.0f : comp;
  }
  __syncthreads();

  const int m       = lane & 15;
  const int koff    = (lane >> 4) * 2;
  const int n       = lane & 15;
  const int colbase = wave * 16;

  v8f acc[MT] = {};
  for (int k = 0; k < D; k += 4) {
    v2f bf, bb;
    bf.x = wf[(size_t)(k + koff)     * D + colbase + n];
    bf.y = wf[(size_t)(k + koff + 1) * D + colbase + n];
    bb.x = wb[(size_t)(k + koff)     * D + colbase + n];
    bb.y = wb[(size_t)(k + koff + 1) * D + colbase + n];
    #pragma unroll
    for (int mt = 0; mt < MT; ++mt) {
      v2f a;
      a.x = cf[(mt * 16 + m) * PITCH + k + koff];
      a.y = cf[(mt * 16 + m) * PITCH + k + koff + 1];
      acc[mt] = __builtin_amdgcn_wmma_f32_16x16x4_f32(false, a, false, bf,
                                                      (short)0, acc[mt], false, false);
      a.x = cb[(mt * 16 + m) * PITCH + k + koff];
      a.y = cb[(mt * 16 + m) * PITCH + k + koff + 1];
      acc[mt] = __builtin_amdgcn_wmma_f32_16x16x4_f32(false, a, false, bb,
                                                      (short)0, acc[mt], false, false);
    }
  }

  // scale by edge_norm (row = edge) and scatter-add into out[dst]
  #pragma unroll
  for (int mt = 0; mt < MT; ++mt) {
    #pragma unroll
    for (int v = 0; v < 8; ++v) {
      int m2 = mt * 16 + v + (lane >> 4) * 8;
      float val = acc[mt][v] * s_norm[m2];
      atomicAdd(&out[(size_t)s_dst[m2] * D + colbase + n], val);
    }
  }
}

// ---------------------------------------------------------------------------
extern "C" void kernel_launch(void* const* d_in, const int* in_sizes, int n_in,
                              void* d_out, int out_size, void* d_ws, size_t ws_size,
                              hipStream_t stream) {
  const float* x    = (const float*)d_in[0];   // [N,128]
  const float* wl   = (const float*)d_in[1];   // [128,128]
  const float* wfwd = (const float*)d_in[2];   // [128,128]
  const float* wbwd = (const float*)d_in[3];   // [128,128]
  const float* rel  = (const float*)d_in[4];   // [100,128]
  const float* en   = (const float*)d_in[5];   // [E]
  const int*   eidx = (const int*)d_in[6];     // [2,E]
  const int*   ety  = (const int*)d_in[7];     // [E]
  float* out = (float*)d_out;

  const int N = in_sizes[0] / D;
  const int E = in_sizes[5];
  const int ROWS = MT * 16;

  // 1) out = x @ w_loop  (writes every element of d_out -> init)
  selfloop_gemm<<<(N + ROWS - 1) / ROWS, 256, 0, stream>>>(x, wl, out, N);
  // 2) scatter-add edge messages on top
  edge_msg<<<(E + ROWS - 1) / ROWS, 256, 0, stream>>>(x, wfwd, wbwd, rel, en, eidx, ety, out, E);
}